// ContrastiveLoss_52321291600468
// MI455X (gfx1250) — compile-verified
//
#include <hip/hip_runtime.h>
#include <hip/hip_bf16.h>
#include <math.h>

// ---------------- problem constants (from reference) ----------------
#define B_    8
#define C_    64
#define H_    200
#define W_    200
#define HW_   (H_ * W_)
#define GSTEP 8          // gridstep = H // 25
#define HM_   24         // len(range(0, 192, 8))
#define NN_   576        // HM*HM
#define NEG_  256
#define KTOT_ 257        // 1 positive + 256 negatives
#define TEMPERATURE_ 2.0f
#define FACTOR_      0.8f

typedef __attribute__((ext_vector_type(2))) float v2f;
typedef __attribute__((ext_vector_type(8))) float v8f;

// ---------------- workspace layout (floats) ----------------
static constexpr size_t Z1_OFF   = 0;                                   // [B][N][C]
static constexpr size_t NA2_OFF  = Z1_OFF   + (size_t)B_ * NN_ * C_;    // [B][N]
static constexpr size_t IMGF_OFF = NA2_OFF  + (size_t)B_ * NN_;         // [B][N][3]
static constexpr size_t SIMK_OFF = IMGF_OFF + (size_t)B_ * NN_ * 3;     // [B][KTOT][N]
static constexpr size_t SIMV_OFF = SIMK_OFF + (size_t)B_ * KTOT_ * NN_; // [B][KTOT]
static constexpr size_t V2T_OFF  = SIMV_OFF + (size_t)B_ * KTOT_;       // [B][H][W][C]
static constexpr size_t WS_BASE_FLOATS = V2T_OFF;                       // ~1.50M (6 MB)
static constexpr size_t WS_FULL_FLOATS = V2T_OFF + (size_t)B_ * HW_ * C_; // ~22.0M (88 MB)

// =====================================================================
// Phase 0 (optional): transpose views_2 [B][C][H][W] -> v2T [B][HW][C]
// so every negative's 64 channels are one contiguous 256B span.
// Block: 256 threads, 64 consecutive pixels x 64 channels via LDS tile.
// grid = B*HW/64 = 5000 blocks.
// =====================================================================
__global__ __launch_bounds__(256) void phase0_transpose(
    const float* __restrict__ v2, float* __restrict__ v2T)
{
    __shared__ float tile[64][65];      // +1 pad: no bank conflicts
    const int blk  = blockIdx.x;        // 0 .. B*HW/64-1
    const int b    = blk / (HW_ / 64);
    const int pos0 = (blk % (HW_ / 64)) * 64;
    const int tid  = threadIdx.x;

    const float* src = v2 + (size_t)b * C_ * HW_ + pos0;
#pragma unroll
    for (int i = 0; i < 16; ++i) {      // 4096 elems / 256 threads
        const int idx = i * 256 + tid;
        const int c   = idx >> 6;       // channel
        const int p   = idx & 63;       // pixel within tile
        // coalesced: 64 consecutive pixels of one channel; NT: v2 not reused
        tile[p][c] = __builtin_nontemporal_load(src + (size_t)c * HW_ + p);
    }
    __syncthreads();

    float* dst = v2T + ((size_t)b * HW_ + pos0) * C_;
#pragma unroll
    for (int i = 0; i < 16; ++i) {
        const int idx = i * 256 + tid;
        const int p   = idx >> 6;
        const int c   = idx & 63;
        dst[(size_t)p * C_ + c] = tile[p][c];   // coalesced, stays in L2
    }
}

// =====================================================================
// Phase 1: gather z1[b,n,c], its squared norm, and anchor RGB (img[0]!)
// grid = B*N blocks, 64 threads (one per channel)
// =====================================================================
__global__ __launch_bounds__(64) void phase1_gather(
    const float* __restrict__ v1, const float* __restrict__ img,
    const int* __restrict__ start_ij,
    float* __restrict__ z1, float* __restrict__ na2, float* __restrict__ imgf)
{
    const int bn = blockIdx.x;
    const int b  = bn / NN_, n = bn % NN_;
    const int c  = threadIdx.x;

    const int s0 = start_ij[b * 2 + 0];
    const int s1 = start_ij[b * 2 + 1];
    const int ii = (n / HM_) * GSTEP + s0;
    const int jj = (n % HM_) * GSTEP + s1;

    const float v = v1[((size_t)b * C_ + c) * HW_ + (size_t)ii * W_ + jj];
    z1[(size_t)bn * C_ + c] = v;

    __shared__ float red[64];
    red[c] = v * v;
    __syncthreads();
#pragma unroll
    for (int s = 32; s > 0; s >>= 1) {
        if (c < s) red[c] += red[c + s];
        __syncthreads();
    }
    if (c == 0) na2[bn] = red[0];

    // reference uses img[0] for every sample (quirk preserved)
    if (c < 3) imgf[(size_t)bn * 3 + c] = img[(size_t)c * HW_ + (size_t)ii * W_ + jj];
}

// =====================================================================
// Phase 2: per (b,n): 256 weighted cosine sims via V_WMMA_F32_16X16X4_F32.
// One wave32 per block. A = z1 row broadcast to 16 rows; B = 16 gathered
// negative columns; accumulate K=64 as 16 chained WMMAs per tile.
// TRANS=true : gather from channel-contiguous v2T (one b64 per step/lane)
// TRANS=false: gather from original layout (two strided b32 per step/lane)
// grid = B*N blocks, 32 threads
// =====================================================================
template <bool TRANS>
__global__ __launch_bounds__(32) void phase2_wmma(
    const float* __restrict__ v2, const float* __restrict__ img,
    const int* __restrict__ start_ij, const int* __restrict__ neg_idx,
    const float* __restrict__ z1, const float* __restrict__ na2,
    const float* __restrict__ imgf, float* __restrict__ simK)
{
    const int bn   = blockIdx.x;
    const int b    = bn / NN_, n = bn % NN_;
    const int lane = threadIdx.x;
    const int half = (lane >> 4) & 1;   // lanes 0-15: K%4 in {0,1}; 16-31: {2,3}
    const int ln   = lane & 15;
    const int cofs = half * 2;

    // A fragments: every M-row identical (broadcast z1 row). f32 16x4 layout:
    // VGPR0 = K=cofs, VGPR1 = K=cofs+1 -> one 8-byte load per step.
    const float* z = z1 + (size_t)bn * C_;
    v2f a[16];
#pragma unroll
    for (int t = 0; t < 16; ++t)
        a[t] = *(const v2f*)(z + 4 * t + cofs);

    const float na2v  = na2[bn];
    const float naval = sqrtf(na2v);

    const int s0 = start_ij[b * 2 + 0];
    const int s1 = start_ij[b * 2 + 1];
    const int ii = (n / HM_) * GSTEP + s0;
    const int jj = (n % HM_) * GSTEP + s1;
    const float if0 = imgf[(size_t)bn * 3 + 0];
    const float if1 = imgf[(size_t)bn * 3 + 1];
    const float if2 = imgf[(size_t)bn * 3 + 2];

    const float inv_max_euc =
        1.0f / sqrtf((float)((H_ - 1) * (H_ - 1) + (W_ - 1) * (W_ - 1)));
    const float inv_max_rgb = 0.57735026919f; // 1/sqrt(3)

    // k == 0 (self-similarity): dot = ||z||^2, na*nb = ||z||^2
    if (lane == 0) {
        const float sim0 = fminf(fabsf(na2v / fmaxf(na2v, 1e-8f)), 1.0f);
        simK[((size_t)b * KTOT_ + 0) * NN_ + n] = sim0;
    }

    const int* negR = neg_idx + ((size_t)(b * 2 + 0) * NN_ + n) * NEG_;
    const int* negC = neg_idx + ((size_t)(b * 2 + 1) * NN_ + n) * NEG_;
    const float* v2b = TRANS ? (v2 + (size_t)b * HW_ * C_)
                             : (v2 + (size_t)b * C_ * HW_);

    for (int tk = 0; tk < 16; ++tk) {
        const int jn = tk * 16 + ln;     // negative index handled by this lane pair
        const int r  = negR[jn];
        const int cl = negC[jn];
        // TRANS: contiguous 64-channel span; else channel-strided plane
        const float* pv = TRANS ? (v2b + ((size_t)r * W_ + cl) * C_)
                                : (v2b + (size_t)r * W_ + cl);

        v8f acc = {0.f, 0.f, 0.f, 0.f, 0.f, 0.f, 0.f, 0.f};
        float ss = 0.0f;
#pragma unroll
        for (int t = 0; t < 16; ++t) {
            const int c0 = 4 * t + cofs;
            v2f bb;
            if (TRANS) {
                bb = *(const v2f*)(pv + c0);             // global_load_b64, 8B aligned
            } else {
                bb[0] = pv[(size_t)c0 * HW_];
                bb[1] = pv[(size_t)(c0 + 1) * HW_];
            }
            ss = fmaf(bb[0], bb[0], fmaf(bb[1], bb[1], ss));
            acc = __builtin_amdgcn_wmma_f32_16x16x4_f32(
                /*neg_a=*/false, a[t], /*neg_b=*/false, bb,
                /*c_mod=*/(short)0, acc, /*reuse_a=*/false, /*reuse_b=*/false);
        }

        // ||neg||^2 = this half's channels + other half's channels
        const float nb2 = ss + __shfl_xor(ss, 16, 32);
        const float dot = acc[0];        // D row M=0 lives in VGPR0, lanes 0-15

        if (lane < 16) {
            const float nb  = sqrtf(nb2);
            const float sim = dot / fmaxf(naval * nb, 1e-8f);

            const float di = (float)(ii - r), dj = (float)(jj - cl);
            const float euc = sqrtf(di * di + dj * dj) * inv_max_euc;

            const float d0 = if0 - img[(size_t)0 * HW_ + (size_t)r * W_ + cl];
            const float d1 = if1 - img[(size_t)1 * HW_ + (size_t)r * W_ + cl];
            const float d2 = if2 - img[(size_t)2 * HW_ + (size_t)r * W_ + cl];
            const float rgb = sqrtf(d0 * d0 + d1 * d1 + d2 * d2) * inv_max_rgb;

            const float wgt = FACTOR_ * euc + (1.0f - FACTOR_) * rgb;
            const float s   = fminf(fabsf(sim * wgt), 1.0f);
            simK[((size_t)b * KTOT_ + 1 + jn) * NN_ + n] = s;
        }
    }
}

// =====================================================================
// Phase 3a: S[b][k] = sum_n simK / N ; k>=1 divided by TEMPERATURE
// grid = B*KTOT blocks, 64 threads
// =====================================================================
__global__ __launch_bounds__(64) void phase3a_reduce_n(
    const float* __restrict__ simK, float* __restrict__ simvec)
{
    const int bk  = blockIdx.x;
    const int tid = threadIdx.x;
    const float* p = simK + (size_t)bk * NN_;

    float s = 0.0f;
    for (int i = tid; i < NN_; i += 64) s += p[i];

    __shared__ float red[64];
    red[tid] = s;
    __syncthreads();
#pragma unroll
    for (int w = 32; w > 0; w >>= 1) {
        if (tid < w) red[tid] += red[tid + w];
        __syncthreads();
    }
    if (tid == 0) {
        float sv = red[0] / (float)NN_;
        if ((bk % KTOT_) > 0) sv /= TEMPERATURE_;
        simvec[bk] = sv;
    }
}

// =====================================================================
// Phase 3b: final 3 scalars. One block, fixed-order tree reduce.
// =====================================================================
__global__ __launch_bounds__(256) void phase3b_final(
    const float* __restrict__ simvec, float* __restrict__ out)
{
    const int tid = threadIdx.x;
    float acc_bce = 0.0f, acc_pos = 0.0f, acc_neg = 0.0f;

    for (int idx = tid; idx < B_ * KTOT_; idx += 256) {
        const int k   = idx % KTOT_;
        const float s = simvec[idx];
        if (k == 0) {
            acc_bce += fmaxf(logf(s), -100.0f);
            acc_pos += s;
        } else {
            acc_bce += fmaxf(logf(1.0f - s), -100.0f);
            acc_neg += s;
        }
    }

    __shared__ float rb[256], rp[256], rn[256];
    rb[tid] = acc_bce; rp[tid] = acc_pos; rn[tid] = acc_neg;
    __syncthreads();
#pragma unroll
    for (int w = 128; w > 0; w >>= 1) {
        if (tid < w) {
            rb[tid] += rb[tid + w];
            rp[tid] += rp[tid + w];
            rn[tid] += rn[tid + w];
        }
        __syncthreads();
    }
    if (tid == 0) {
        out[0] = -rb[0] / (float)(B_ * KTOT_);                       // loss
        out[1] = rp[0] / (float)B_;                                  // sim_all[0]/B
        out[2] = rn[0] / (float)NEG_ * TEMPERATURE_ / (float)B_;     // neg term
    }
}

// =====================================================================
extern "C" void kernel_launch(void* const* d_in, const int* in_sizes, int n_in,
                              void* d_out, int out_size, void* d_ws, size_t ws_size,
                              hipStream_t stream)
{
    const float* views_1  = (const float*)d_in[0];
    const float* views_2  = (const float*)d_in[1];
    const float* img      = (const float*)d_in[2];
    const int*   start_ij = (const int*)d_in[3];
    const int*   neg_idx  = (const int*)d_in[4];
    float*       out      = (float*)d_out;

    float* ws     = (float*)d_ws;
    float* z1     = ws + Z1_OFF;
    float* na2    = ws + NA2_OFF;
    float* imgf   = ws + IMGF_OFF;
    float* simK   = ws + SIMK_OFF;
    float* simvec = ws + SIMV_OFF;
    float* v2T    = ws + V2T_OFF;

    // Transposed (channel-contiguous) gather path needs ~88 MB of scratch;
    // fall back to direct strided gathers if the workspace is smaller.
    const bool use_t = ws_size >= WS_FULL_FLOATS * sizeof(float);

    phase1_gather<<<B_ * NN_, 64, 0, stream>>>(views_1, img, start_ij,
                                               z1, na2, imgf);
    if (use_t) {
        phase0_transpose<<<B_ * (HW_ / 64), 256, 0, stream>>>(views_2, v2T);
        phase2_wmma<true><<<B_ * NN_, 32, 0, stream>>>(v2T, img, start_ij, neg_idx,
                                                       z1, na2, imgf, simK);
    } else {
        phase2_wmma<false><<<B_ * NN_, 32, 0, stream>>>(views_2, img, start_ij, neg_idx,
                                                        z1, na2, imgf, simK);
    }
    phase3a_reduce_n<<<B_ * KTOT_, 64, 0, stream>>>(simK, simvec);
    phase3b_final<<<1, 256, 0, stream>>>(simvec, out);
}